// GAT_6098853560688
// MI455X (gfx1250) — compile-verified
//
#include <hip/hip_runtime.h>
#include <hip/hip_bf16.h>

typedef __attribute__((ext_vector_type(2))) float v2f;
typedef __attribute__((ext_vector_type(8))) float v8f;

#define NEG_SLOPE 0.2f
#define NEG_FLT_MAX (-3.402823466e+38f)

// ---- float atomic max via ordered-int trick (init with -FLT_MAX) ----
__device__ __forceinline__ void atomicMaxFloat(float* addr, float val) {
    if (val >= 0.0f) atomicMax((int*)addr, __float_as_int(val));
    else             atomicMin((unsigned int*)addr, __float_as_uint(val));
}

// ============================ init kernels ============================
__global__ void k_fill(float* __restrict__ p, float v, int n) {
    int i = blockIdx.x * blockDim.x + threadIdx.x;
    if (i < n) p[i] = v;
}
__global__ void k_fill_from(float* __restrict__ p, const float* __restrict__ v, int n) {
    int i = blockIdx.x * blockDim.x + threadIdx.x;
    if (i < n) p[i] = v[0];
}

// ================= layer-1 GEMM: H[N,128] = X[N,128] @ W[128,128] =================
// Block = 256 threads = 8 waves. Block stages a 64x128 A tile (32 KB) in LDS once;
// wave w owns column tile [16w,16w+16) and computes 4 row sub-tiles, reusing each
// B fetch across 4 WMMAs. fp32 WMMA 16x16x4, 32 K-steps.
__global__ __launch_bounds__(256) void k_gemm_wmma(const float* __restrict__ X,
                                                   const float* __restrict__ W,
                                                   float* __restrict__ Hout, int N) {
    __shared__ float As[64 * 128];               // 32 KB of the 320 KB/WGP LDS
    const int tid  = threadIdx.x;
    const int lane = tid & 31;
    const int wave = tid >> 5;                   // column tile 0..7
    const int g    = lane >> 4;                  // half-wave group
    const int m    = lane & 15;
    const int row0 = blockIdx.x * 64;

    // ---- stage 64x128 A tile to LDS (coalesced float4, clamped) ----
    {
        const float4* xg = (const float4*)X;
        float4*       sl = (float4*)As;
        const long long maxv = (long long)N * 32 - 1;   // last valid float4 of X
        #pragma unroll
        for (int i = 0; i < 8; ++i) {
            long long gidx = (long long)row0 * 32 + tid + i * 256;
            if (gidx > maxv) gidx = maxv;
            sl[tid + i * 256] = xg[gidx];
        }
    }
    __syncthreads();

    const int colBase = wave * 16;
    v8f acc0 = {}, acc1 = {}, acc2 = {}, acc3 = {};
    for (int k0 = 0; k0 < 128; k0 += 4) {
        const int k = k0 + 2 * g;
        // B 4x16: lane holds col m, rows {k, k+1}; fetched once, used 4x
        v2f b;
        b[0] = W[(size_t)k * 128 + colBase + m];
        b[1] = W[(size_t)(k + 1) * 128 + colBase + m];
        // A 16x4 per sub-tile: lane holds row m, K = {k, k+1} (from LDS)
        v2f a0 = *(const v2f*)(As + (0 * 16 + m) * 128 + k);
        v2f a1 = *(const v2f*)(As + (1 * 16 + m) * 128 + k);
        v2f a2 = *(const v2f*)(As + (2 * 16 + m) * 128 + k);
        v2f a3 = *(const v2f*)(As + (3 * 16 + m) * 128 + k);
        acc0 = __builtin_amdgcn_wmma_f32_16x16x4_f32(false, a0, false, b, (short)0, acc0, false, false);
        acc1 = __builtin_amdgcn_wmma_f32_16x16x4_f32(false, a1, false, b, (short)0, acc1, false, false);
        acc2 = __builtin_amdgcn_wmma_f32_16x16x4_f32(false, a2, false, b, (short)0, acc2, false, false);
        acc3 = __builtin_amdgcn_wmma_f32_16x16x4_f32(false, a3, false, b, (short)0, acc3, false, false);
    }

    // D 16x16 f32: VGPR v -> row v + 8*g, col m
    #pragma unroll
    for (int rt = 0; rt < 4; ++rt) {
        v8f acc = (rt == 0) ? acc0 : (rt == 1) ? acc1 : (rt == 2) ? acc2 : acc3;
        #pragma unroll
        for (int v = 0; v < 8; ++v) {
            int r = row0 + rt * 16 + v + 8 * g;
            if (r < N) Hout[(size_t)r * 128 + colBase + m] = acc[v];
        }
    }
}

// ================= per-node attention logits: a_s/a_d [N,4] =================
__global__ void k_att_nh(const float* __restrict__ H1, const float* __restrict__ att_s,
                         const float* __restrict__ att_d, float* __restrict__ as_,
                         float* __restrict__ ad_, int N) {
    int t = blockIdx.x * blockDim.x + threadIdx.x;
    if (t >= N * 4) return;
    int n = t >> 2, hh = t & 3;
    const float4* hp = (const float4*)(H1 + (size_t)n * 128 + hh * 32);
    const float4* sp = (const float4*)(att_s + hh * 32);
    const float4* dp = (const float4*)(att_d + hh * 32);
    float s = 0.f, d = 0.f;
    for (int i = 0; i < 8; ++i) {
        float4 h = hp[i], a = sp[i], b = dp[i];
        s += h.x * a.x + h.y * a.y + h.z * a.z + h.w * a.w;
        d += h.x * b.x + h.y * b.y + h.z * b.z + h.w * b.w;
    }
    as_[t] = s; ad_[t] = d;
}

// ============= edge pass 1 (per edge x head): leaky-relu score + segment max ============
__device__ __forceinline__ void edge_sd(const long long* ei, int E, int e,
                                        long long& s, long long& d) {
    if (e < E) { s = ei[e]; d = ei[(size_t)E + e]; }
    else       { s = d = (long long)(e - E); }      // appended self-loops
}

__global__ void k_edge_score(const long long* __restrict__ ei, int E, int Etot,
                             const float* __restrict__ as_, const float* __restrict__ ad_,
                             float* __restrict__ esc, float* __restrict__ mx) {
    int t = blockIdx.x * blockDim.x + threadIdx.x;
    if (t >= Etot * 4) return;
    int e = t >> 2, hh = t & 3;
    long long s, d; edge_sd(ei, E, e, s, d);
    float v = as_[s * 4 + hh] + ad_[d * 4 + hh];
    v = (v > 0.f) ? v : NEG_SLOPE * v;
    esc[t] = v;
    atomicMaxFloat(&mx[d * 4 + hh], v);
}

__global__ void k_edge_exp(const long long* __restrict__ ei, int E, int Etot,
                           const float* __restrict__ mx, float* __restrict__ esc,
                           float* __restrict__ den) {
    int t = blockIdx.x * blockDim.x + threadIdx.x;
    if (t >= Etot * 4) return;
    int e = t >> 2, hh = t & 3;
    long long s, d; edge_sd(ei, E, e, s, d); (void)s;
    float p = __expf(esc[t] - mx[d * 4 + hh]);
    esc[t] = p;                               // in-place: esc now holds p
    atomicAdd(&den[d * 4 + hh], p);
}

// ============ message aggregation: acc[dst,:] += alpha * h[src,:] (coalesced) ============
__global__ __launch_bounds__(256) void k_aggregate(const long long* __restrict__ ei, int E,
                                                   int Etot, const float* __restrict__ p,
                                                   const float* __restrict__ den,
                                                   const float* __restrict__ H1,
                                                   float* __restrict__ acc) {
    int e = blockIdx.x * 2 + (threadIdx.x >> 7);   // 2 edges per 256-thread block
    if (e >= Etot) return;
    int i = threadIdx.x & 127;                     // channel 0..127
    long long s, d; edge_sd(ei, E, e, s, d);
    int hh = i >> 5;
    float alpha = p[(size_t)e * 4 + hh] / den[d * 4 + hh];
    float val = alpha * H1[(size_t)s * 128 + i];
    atomicAdd(&acc[(size_t)d * 128 + i], val);
}

// ===================== bias + ELU (in place) =====================
__global__ void k_bias_elu(float* __restrict__ acc, const float* __restrict__ bias, int total) {
    int t = blockIdx.x * blockDim.x + threadIdx.x;
    if (t >= total) return;
    float v = acc[t] + bias[t & 127];
    acc[t] = (v > 0.f) ? v : (__expf(v) - 1.0f);
}

// ===================== layer-2 projection (128 -> 1) =====================
__global__ void k_proj2(const float* __restrict__ H2, const float* __restrict__ W2,
                        const float* __restrict__ asrc, const float* __restrict__ adst,
                        float* __restrict__ z, float* __restrict__ as2,
                        float* __restrict__ ad2, int N) {
    int n = blockIdx.x * blockDim.x + threadIdx.x;
    if (n >= N) return;
    const float4* hp = (const float4*)(H2 + (size_t)n * 128);
    const float4* wp = (const float4*)W2;
    float acc = 0.f;
    for (int i = 0; i < 32; ++i) {
        float4 h = hp[i], w = wp[i];
        acc += h.x * w.x + h.y * w.y + h.z * w.z + h.w * w.w;
    }
    z[n] = acc;
    as2[n] = acc * asrc[0];
    ad2[n] = acc * adst[0];
}

// ===================== layer-2 scalar edge softmax + aggregate =====================
__global__ void k_edge_score_s(const long long* __restrict__ ei, int E, int Etot,
                               const float* __restrict__ as2, const float* __restrict__ ad2,
                               float* __restrict__ e2, float* __restrict__ m2) {
    int e = blockIdx.x * blockDim.x + threadIdx.x;
    if (e >= Etot) return;
    long long s, d; edge_sd(ei, E, e, s, d);
    float v = as2[s] + ad2[d];
    v = (v > 0.f) ? v : NEG_SLOPE * v;
    e2[e] = v;
    atomicMaxFloat(&m2[d], v);
}
__global__ void k_edge_exp_s(const long long* __restrict__ ei, int E, int Etot,
                             const float* __restrict__ m2, float* __restrict__ e2,
                             float* __restrict__ den2) {
    int e = blockIdx.x * blockDim.x + threadIdx.x;
    if (e >= Etot) return;
    long long s, d; edge_sd(ei, E, e, s, d); (void)s;
    float p = __expf(e2[e] - m2[d]);
    e2[e] = p;
    atomicAdd(&den2[d], p);
}
__global__ void k_edge_agg_s(const long long* __restrict__ ei, int E, int Etot,
                             const float* __restrict__ p2, const float* __restrict__ den2,
                             const float* __restrict__ z, float* __restrict__ out) {
    int e = blockIdx.x * blockDim.x + threadIdx.x;
    if (e >= Etot) return;
    long long s, d; edge_sd(ei, E, e, s, d);
    float w = p2[e] / den2[d];
    atomicAdd(&out[d], w * z[s]);
}

// ============================ launcher ============================
extern "C" void kernel_launch(void* const* d_in, const int* in_sizes, int n_in,
                              void* d_out, int out_size, void* d_ws, size_t ws_size,
                              hipStream_t stream) {
    const float*     x     = (const float*)d_in[0];
    const long long* ei    = (const long long*)d_in[1];   // int64 [2,E]
    const float*     W1    = (const float*)d_in[2];
    const float*     asrc1 = (const float*)d_in[3];
    const float*     adst1 = (const float*)d_in[4];
    const float*     bias1 = (const float*)d_in[5];
    const float*     W2    = (const float*)d_in[6];
    const float*     asrc2 = (const float*)d_in[7];
    const float*     adst2 = (const float*)d_in[8];
    const float*     bias2 = (const float*)d_in[9];
    float*           out   = (float*)d_out;

    const int N    = in_sizes[0] / 128;
    const int E    = in_sizes[1] / 2;
    const int Etot = E + N;       // with self-loops

    // ---- workspace layout (floats) ----
    float* ws = (float*)d_ws;
    size_t off = 0;
    float* h1   = ws + off; off += (size_t)N * 128;
    float* as1  = ws + off; off += (size_t)N * 4;
    float* ad1  = ws + off; off += (size_t)N * 4;
    float* m1   = ws + off; off += (size_t)N * 4;
    float* den1 = ws + off; off += (size_t)N * 4;
    float* esc1 = ws + off; off += (size_t)Etot * 4;   // scores -> probs (in place)
    float* acc  = ws + off; off += (size_t)N * 128;    // layer-1 out -> h2 (in place)
    float* z    = ws + off; off += (size_t)N;
    float* as2  = ws + off; off += (size_t)N;
    float* ad2  = ws + off; off += (size_t)N;
    float* m2   = ws + off; off += (size_t)N;
    float* den2 = ws + off; off += (size_t)N;
    float* e2   = ws + off; off += (size_t)Etot;
    (void)ws_size; (void)n_in; (void)out_size;

    const int T = 256;
    auto blk = [](long long n, int t) { return (unsigned)((n + t - 1) / t); };

    // ---- init (every call: buffers are not re-poisoned between replays) ----
    k_fill<<<blk((long long)N * 4, T), T, 0, stream>>>(m1, NEG_FLT_MAX, N * 4);
    k_fill<<<blk((long long)N * 4, T), T, 0, stream>>>(den1, 0.f, N * 4);
    k_fill<<<blk((long long)N * 128, T), T, 0, stream>>>(acc, 0.f, N * 128);
    k_fill<<<blk(N, T), T, 0, stream>>>(m2, NEG_FLT_MAX, N);
    k_fill<<<blk(N, T), T, 0, stream>>>(den2, 0.f, N);
    k_fill_from<<<blk(N, T), T, 0, stream>>>(out, bias2, N);   // out starts at bias2

    // ---- layer 1 ----
    k_gemm_wmma<<<(N + 63) / 64, 256, 0, stream>>>(x, W1, h1, N);
    k_att_nh<<<blk((long long)N * 4, T), T, 0, stream>>>(h1, asrc1, adst1, as1, ad1, N);
    k_edge_score<<<blk((long long)Etot * 4, T), T, 0, stream>>>(ei, E, Etot, as1, ad1, esc1, m1);
    k_edge_exp<<<blk((long long)Etot * 4, T), T, 0, stream>>>(ei, E, Etot, m1, esc1, den1);
    k_aggregate<<<blk(Etot, 2), 256, 0, stream>>>(ei, E, Etot, esc1, den1, h1, acc);
    k_bias_elu<<<blk((long long)N * 128, T), T, 0, stream>>>(acc, bias1, N * 128);

    // ---- layer 2 (H=C=1) ----
    k_proj2<<<blk(N, T), T, 0, stream>>>(acc, W2, asrc2, adst2, z, as2, ad2, N);
    k_edge_score_s<<<blk(Etot, T), T, 0, stream>>>(ei, E, Etot, as2, ad2, e2, m2);
    k_edge_exp_s<<<blk(Etot, T), T, 0, stream>>>(ei, E, Etot, m2, e2, den2);
    k_edge_agg_s<<<blk(Etot, T), T, 0, stream>>>(ei, E, Etot, e2, den2, z, out);
}